// RelativeLoss_95_6605659701729
// MI455X (gfx1250) — compile-verified
//
#include <hip/hip_runtime.h>
#include <stdint.h>

// ---------------------------------------------------------------------------
// Trimmed-mean of squared relative error via 2-pass radix-histogram selection.
//   pass1: 32768-bin LDS histogram of float bit-pattern >> 16 (monotone for
//          err >= 0).  128KB LDS per 1024-thread block (CDNA5: 320KB/WGP).
//   select: scan bins -> threshold bin B, needed = k - count_below.
//   pass2: sum err for bin<B, sum+count for bin==B.  Wave reduction uses
//          V_WMMA_F32_16X16X4_F32 (A = lane values in K=0, B = ones).
//   final: mean = (sumBelow + needed * tieAvg) / k.
// ---------------------------------------------------------------------------

#define RL_NBINS 32768
#define RL_NB    256      // blocks for streaming passes
#define RL_NT    1024     // threads per block (32 waves of 32)

typedef float rl_v2f __attribute__((ext_vector_type(2)));
typedef float rl_v8f __attribute__((ext_vector_type(8)));

// Exact wave32 sum via V_WMMA_F32_16X16X4_F32.
// A(16x4): lane m (0..15) VGPR0 -> A[m][0], lanes 16..31 VGPR0 -> A[m][2].
// With a = {v, 0} and B = all ones: D[m][n] = v_m + v_{m+16} for all n.
// C/D layout: lane n VGPR r = D[r][n] (lanes 0..15), D[r+8][n] (lanes 16..31),
// so sum of the 8 C VGPRs gives half-sums; one xor-16 shuffle completes it.
__device__ __forceinline__ float rl_wave_sum(float v) {
  rl_v2f a; a.x = v;    a.y = 0.0f;
  rl_v2f b; b.x = 1.0f; b.y = 1.0f;
  rl_v8f c = {0.f, 0.f, 0.f, 0.f, 0.f, 0.f, 0.f, 0.f};
  c = __builtin_amdgcn_wmma_f32_16x16x4_f32(false, a, false, b,
                                            (short)0, c, false, false);
  float t = c[0] + c[1] + c[2] + c[3] + c[4] + c[5] + c[6] + c[7];
  t += __shfl_xor(t, 16, 32);
  return t;  // full wave sum in every lane
}

__device__ __forceinline__ uint32_t rl_bin(float e) {
  // e >= 0 -> sign bit 0 -> result in [0, 32768)
  return __float_as_uint(e) >> 16;
}

__device__ __forceinline__ float rl_err(float o, float t) {
  float d = (t - o) / t;
  return d * d;
}

// ---- zero workspace --------------------------------------------------------
__global__ void rl_zero_kernel(uint32_t* words, int nwords,
                               double* dbls, int ndbls) {
  int i = blockIdx.x * blockDim.x + threadIdx.x;
  if (i < nwords) words[i] = 0u;
  if (i < ndbls)  dbls[i]  = 0.0;
}

// ---- pass 1: histogram -----------------------------------------------------
__global__ void __launch_bounds__(RL_NT)
rl_hist_kernel(const float* __restrict__ out, const float* __restrict__ tgt,
               uint32_t* __restrict__ ghist, int n) {
  extern __shared__ uint32_t lhist[];  // RL_NBINS entries (128 KB dynamic LDS)
  for (int j = threadIdx.x; j < RL_NBINS; j += blockDim.x) lhist[j] = 0u;
  __syncthreads();

  const int n4     = n >> 2;
  const int tid    = blockIdx.x * blockDim.x + threadIdx.x;
  const int stride = gridDim.x * blockDim.x;
  const float4* o4 = (const float4*)out;
  const float4* t4 = (const float4*)tgt;

  for (int i = tid; i < n4; i += stride) {
    __builtin_prefetch(&o4[i + stride], 0, 1);   // global_prefetch_b8
    __builtin_prefetch(&t4[i + stride], 0, 1);
    float4 o = o4[i];
    float4 t = t4[i];
    atomicAdd(&lhist[rl_bin(rl_err(o.x, t.x))], 1u);
    atomicAdd(&lhist[rl_bin(rl_err(o.y, t.y))], 1u);
    atomicAdd(&lhist[rl_bin(rl_err(o.z, t.z))], 1u);
    atomicAdd(&lhist[rl_bin(rl_err(o.w, t.w))], 1u);
  }
  int rem = n & 3;
  if (tid < rem) {
    int i = (n4 << 2) + tid;
    atomicAdd(&lhist[rl_bin(rl_err(out[i], tgt[i]))], 1u);
  }
  __syncthreads();

  for (int j = threadIdx.x; j < RL_NBINS; j += blockDim.x) {
    uint32_t v = lhist[j];
    if (v) atomicAdd(&ghist[j], v);
  }
}

// ---- select threshold bin --------------------------------------------------
__global__ void __launch_bounds__(RL_NT)
rl_select_kernel(const uint32_t* __restrict__ ghist,
                 uint32_t* __restrict__ meta, uint32_t k) {
  __shared__ uint32_t part[RL_NT];
  const int per = RL_NBINS / RL_NT;  // 32 bins per thread
  uint32_t s = 0;
  int base = threadIdx.x * per;
  for (int j = 0; j < per; ++j) s += ghist[base + j];
  part[threadIdx.x] = s;
  __syncthreads();

  if (threadIdx.x == 0) {
    uint32_t cum = 0;
    int chunk = 0;
    for (; chunk < RL_NT; ++chunk) {
      if (cum + part[chunk] >= k) break;
      cum += part[chunk];
    }
    uint32_t b = (uint32_t)chunk * per;
    for (;; ++b) {
      uint32_t h = ghist[b];
      if (cum + h >= k) break;
      cum += h;
    }
    meta[0] = b;        // threshold bin B
    meta[1] = k - cum;  // elements needed from bin B (>=1)
  }
}

// ---- pass 2: sums ----------------------------------------------------------
__global__ void __launch_bounds__(RL_NT)
rl_sum_kernel(const float* __restrict__ out, const float* __restrict__ tgt,
              const uint32_t* __restrict__ meta,
              double* __restrict__ partials, int n) {
  const uint32_t B = meta[0];
  float sBelow = 0.0f, sTie = 0.0f, cTie = 0.0f;

  const int n4     = n >> 2;
  const int tid    = blockIdx.x * blockDim.x + threadIdx.x;
  const int stride = gridDim.x * blockDim.x;
  const float4* o4 = (const float4*)out;
  const float4* t4 = (const float4*)tgt;

  for (int i = tid; i < n4; i += stride) {
    __builtin_prefetch(&o4[i + stride], 0, 1);
    __builtin_prefetch(&t4[i + stride], 0, 1);
    float4 o = o4[i];
    float4 t = t4[i];
    float e0 = rl_err(o.x, t.x), e1 = rl_err(o.y, t.y);
    float e2 = rl_err(o.z, t.z), e3 = rl_err(o.w, t.w);
    uint32_t b0 = rl_bin(e0), b1 = rl_bin(e1), b2 = rl_bin(e2), b3 = rl_bin(e3);
    if (b0 < B) sBelow += e0; else if (b0 == B) { sTie += e0; cTie += 1.0f; }
    if (b1 < B) sBelow += e1; else if (b1 == B) { sTie += e1; cTie += 1.0f; }
    if (b2 < B) sBelow += e2; else if (b2 == B) { sTie += e2; cTie += 1.0f; }
    if (b3 < B) sBelow += e3; else if (b3 == B) { sTie += e3; cTie += 1.0f; }
  }
  int rem = n & 3;
  if (tid < rem) {
    int i = (n4 << 2) + tid;
    float e = rl_err(out[i], tgt[i]);
    uint32_t b = rl_bin(e);
    if (b < B) sBelow += e; else if (b == B) { sTie += e; cTie += 1.0f; }
  }

  // wave32 reduction via WMMA, then 32 waves -> block via wave 0
  __shared__ float red[3][32];
  const int lane = threadIdx.x & 31;
  const int wave = threadIdx.x >> 5;

  float w0 = rl_wave_sum(sBelow);
  float w1 = rl_wave_sum(sTie);
  float w2 = rl_wave_sum(cTie);
  if (lane == 0) { red[0][wave] = w0; red[1][wave] = w1; red[2][wave] = w2; }
  __syncthreads();

  if (wave == 0) {
    float a0 = red[0][lane];
    float a1 = red[1][lane];
    float a2 = red[2][lane];
    float t0 = rl_wave_sum(a0);
    float t1 = rl_wave_sum(a1);
    float t2 = rl_wave_sum(a2);
    if (lane == 0) {
      partials[blockIdx.x * 3 + 0] = (double)t0;
      partials[blockIdx.x * 3 + 1] = (double)t1;
      partials[blockIdx.x * 3 + 2] = (double)t2;
    }
  }
}

// ---- finalize (deterministic serial fold of 256 partials) ------------------
__global__ void rl_final_kernel(const double* __restrict__ partials,
                                const uint32_t* __restrict__ meta,
                                float* __restrict__ result, uint32_t k) {
  double sBelow = 0.0, sTie = 0.0, cTie = 0.0;
  for (int i = 0; i < RL_NB; ++i) {
    sBelow += partials[i * 3 + 0];
    sTie   += partials[i * 3 + 1];
    cTie   += partials[i * 3 + 2];
  }
  uint32_t needed = meta[1];
  double tieAvg = (cTie > 0.0) ? (sTie / cTie) : 0.0;
  result[0] = (float)((sBelow + (double)needed * tieAvg) / (double)k);
}

// ---------------------------------------------------------------------------
extern "C" void kernel_launch(void* const* d_in, const int* in_sizes, int n_in,
                              void* d_out, int out_size, void* d_ws, size_t ws_size,
                              hipStream_t stream) {
  const float* d_output = (const float*)d_in[0];
  const float* d_target = (const float*)d_in[1];
  float* d_res = (float*)d_out;
  const int n = in_sizes[0];
  const uint32_t k = (uint32_t)((double)n * 0.97);

  // workspace layout
  uint32_t* ghist    = (uint32_t*)d_ws;                         // 32768 u32
  uint32_t* meta     = ghist + RL_NBINS;                        // 4 u32
  double*   partials = (double*)((char*)d_ws + RL_NBINS * 4 + 16); // 8-aligned, RL_NB*3 f64

  const int nwords = RL_NBINS + 4;
  const int ndbls  = RL_NB * 3;
  int zthreads = (nwords > ndbls) ? nwords : ndbls;
  rl_zero_kernel<<<(zthreads + 255) / 256, 256, 0, stream>>>(ghist, nwords,
                                                             partials, ndbls);

  rl_hist_kernel<<<RL_NB, RL_NT, RL_NBINS * sizeof(uint32_t), stream>>>(
      d_output, d_target, ghist, n);

  rl_select_kernel<<<1, RL_NT, 0, stream>>>(ghist, meta, k);

  rl_sum_kernel<<<RL_NB, RL_NT, 0, stream>>>(d_output, d_target, meta,
                                             partials, n);

  rl_final_kernel<<<1, 1, 0, stream>>>(partials, meta, d_res, k);
}